// HIRNNLayer_1151051235849
// MI455X (gfx1250) — compile-verified
//
#include <hip/hip_runtime.h>
#include <hip/hip_bf16.h>
#include <stdint.h>

// ---------------------------------------------------------------------------
// HIRNN (hydrological bucket RNN) fused scan + Q output for gfx1250.
//   - 1 lane per batch element, 1 wave (32 lanes) per block -> 128 blocks.
//   - Per-block input tile (32 rows x 64 timesteps x float4 = 32KB) staged
//     into LDS by the Tensor Data Mover (tensor_load_to_lds), double-buffered,
//     pipelined with s_wait_tensorcnt (per-wave counter -> no barriers).
//   - TDM LDS padding: 16B pad per 1024B row => 1040B row stride (4-dword
//     bank skew, 16B-aligned ds_load_b128, 2 lanes/bank = optimal).
//   - H(x)=(tanh(1e7 x)+1)/2 is an exact f32 step function; all H-blends that
//     are exact min/max/relu identities (incl. the H(0)=0.5 boundary) are
//     folded to v_min/v_max to shorten the serial dependency chain.
// ---------------------------------------------------------------------------

#define T_LEN 2048
#define TT 64                       // timesteps per tile
#define NT (T_LEN / TT)             // 32 tiles
#define ROW_BYTES (TT * 16)         // 1024 B of data per lane-row
#define ROW_STRIDE (ROW_BYTES + 16) // 1040 B incl. TDM pad
#define BUF_BYTES (32 * ROW_STRIDE) // 33280 B per buffer
#define CLIPV 100000.0f

#ifndef __has_builtin
#define __has_builtin(x) 0
#endif

#if __has_builtin(__builtin_amdgcn_tensor_load_to_lds)
#define HAS_TDM 1
#else
#define HAS_TDM 0
#endif

typedef unsigned int u32;
typedef unsigned long long u64;
typedef u32 uint32x4 __attribute__((ext_vector_type(4)));
typedef int int32x4 __attribute__((ext_vector_type(4)));
typedef int int32x8 __attribute__((ext_vector_type(8)));

// ---- tensorcnt waits -------------------------------------------------------
#if __has_builtin(__builtin_amdgcn_s_wait_tensorcnt)
__device__ __forceinline__ void wait_tc0() { __builtin_amdgcn_s_wait_tensorcnt(0); }
__device__ __forceinline__ void wait_tc1() { __builtin_amdgcn_s_wait_tensorcnt(1); }
#else
__device__ __forceinline__ void wait_tc0() { asm volatile("s_wait_tensorcnt 0x0" ::: "memory"); }
__device__ __forceinline__ void wait_tc1() { asm volatile("s_wait_tensorcnt 0x1" ::: "memory"); }
#endif

// ---- fast math helpers -----------------------------------------------------
__device__ __forceinline__ float fast_rcp(float x) {
#if __has_builtin(__builtin_amdgcn_rcpf)
  return __builtin_amdgcn_rcpf(x);      // v_rcp_f32
#else
  return 1.0f / x;
#endif
}
__device__ __forceinline__ float fast_sqrt(float x) {
#if __has_builtin(__builtin_amdgcn_sqrtf)
  return __builtin_amdgcn_sqrtf(x);     // v_sqrt_f32, no IEEE fixup
#else
  return sqrtf(x);
#endif
}

// ---- model params ----------------------------------------------------------
struct Params {
  float insc, S, invS, C, q, sub, crak, rk, lg, kr, ls;
};

__device__ __forceinline__ float clipf(float x) {
  return fminf(fmaxf(x, -CLIPV), CLIPV);
}
// Exact step function matching tanh(1e7*x) saturation in f32, incl. H(0)=0.5.
// Only needed where the two blend branches differ at x==0 (Q_ir).
__device__ __forceinline__ float Hstep(float x) {
  return (x > 0.0f) ? 1.0f : ((x < 0.0f) ? 0.0f : 0.5f);
}

struct Flux { float IRUN, SRUN, REC, SMF, ETS, BAS, SMS1c; };

__device__ __forceinline__ Flux fluxes(float SMSp, float GWp, float4 in, const Params& p) {
  Flux f;
  float Prec = in.x, PET = in.y;
  float IMAX  = fmaxf(fminf(p.insc, PET), 0.0f);
  float INTm  = fmaxf(fminf(IMAX, Prec), 0.0f);
  float INR   = fmaxf(Prec - INTm, 0.0f);
  float SMS1c = fmaxf(fminf(SMSp, p.S), 0.0f);
  float ratio = SMS1c * p.invS;
  float infil = p.C * __expf(-p.q * ratio);
  // H(infil-INR)*INR + H(INR-infil)*infil == min(INR, infil)  (exact at ==)
  float RMO   = fmaxf(fminf(INR, infil), 0.0f);
  f.IRUN = fmaxf(INR - RMO, 0.0f);
  f.SRUN = fmaxf(p.sub * ratio * RMO, 0.0f);
  f.REC  = fmaxf(p.crak * ratio * (RMO - f.SRUN), 0.0f);
  f.SMF  = fmaxf(RMO - f.SRUN - f.REC, 0.0f);
  float POT = fmaxf(PET - INTm, 0.0f);
  float cap = 10.0f * ratio;
  // H(cap-POT)*POT + H(POT-cap)*cap == min(POT, cap)  (exact at ==)
  f.ETS = fmaxf(fminf(POT, cap), 0.0f);
  // H(GW)*(rk*GW) clamped at 0 == max(rk*GW, 0) since rk > 0 (exact at GW==0)
  f.BAS = fmaxf(p.rk * GWp, 0.0f);
  f.SMS1c = SMS1c;
  return f;
}

__device__ __forceinline__ void release(float RS, float inflow, float RSmax, float rcpRSmax,
                                        const Params& p, float& Q_or, float& Q_ir) {
  float x = RS + inflow - RSmax;
  // H(x)*x clamped at 0 == max(x, 0)
  Q_or = fmaxf(x, 0.0f);
  // Q_ir branches genuinely differ at x==0 -> keep the H blend (H(0)=0.5).
  float hr = Hstep(x);
  float z  = RS * rcpRSmax;                 // RSmax >= 10
  Q_ir = fmaxf(hr * (p.kr * RSmax) + (1.0f - hr) * (p.kr * RS * (z * fast_sqrt(z))), 0.0f);
}

// Output Q for time t: fluxes from pre-step state, RS = POST-step reservoir.
__device__ __forceinline__ float qout(const Flux& f, float RSpost, float4 in,
                                      float rcpRSmax, const Params& p) {
  float Area = in.w, RSmax = in.z;
  float inflow = (f.IRUN + f.SRUN + f.BAS) * Area;
  float Q_or, Q_ir;
  release(RSpost, inflow, RSmax, rcpRSmax, p, Q_or, Q_ir);
  float oneA = 1.0f - Area;
  float dr = (f.SRUN + f.IRUN) * oneA + Q_ir + Q_or - p.ls;
  // H(dr)*dr clamped at 0 == max(dr, 0)
  float DR = fmaxf(dr, 0.0f);
  return fmaxf(DR + f.BAS * oneA, 0.0f);
}

// One fused scan step; advances (SMS,GW,RS) and returns Q[t] (valid for t>=1).
__device__ __forceinline__ float step(float4 in, float& SMS, float& GW, float& RS,
                                      const Params& p) {
  Flux f = fluxes(SMS, GW, in, p);
  float nextSMS = f.SMS1c + clipf(f.SMF - f.ETS);
  float xre = f.SMS1c + f.SMF - f.ETS - p.S;
  // H(x)*(REC+x) + H(-x)*REC == REC + max(x, 0)  (exact at x==0)
  float RECn = fmaxf(f.REC + fmaxf(xre, 0.0f), 0.0f);
  float nextGW = (GW - p.lg) + clipf(RECn - f.BAS);
  float Area = in.w, RSmax = in.z;
  float rcpRSmax = fast_rcp(RSmax);
  float inflow = (f.IRUN + f.SRUN + f.BAS) * Area;
  float Q_or, Q_ir;
  release(RS, inflow, RSmax, rcpRSmax, p, Q_or, Q_ir);   // uses PRE-step RS
  float Qin = fmaxf(inflow, 0.0f);
  float nextRS = RS + clipf(Qin - Q_or - Q_ir);
  float Q = qout(f, nextRS, in, rcpRSmax, p);            // uses POST-step RS
  SMS = nextSMS; GW = nextGW; RS = nextRS;
  return Q;
}

#if HAS_TDM
// 2D TDM descriptor: rows = 32 batch elements, row = TT*4 f32 (contiguous in t),
// row-to-row global stride = T_LEN*4 elements. Pad 4 dwords per 256 dwords
// stored -> 16B pad exactly at each 1024B row boundary.
// 6-arg builtin: (uint32x4 g0, int32x8 g1, int32x4 g2, int32x4 g3, int32x8, i32 cpol)
__device__ __forceinline__ void tdm_load_tile(const float* __restrict__ inputs,
                                              int b0, int t0, u32 ldsOff) {
  u64 ga = (u64)(uintptr_t)(inputs + ((u64)b0 * T_LEN + t0) * 4);
  uint32x4 g0;
  g0[0] = 1u;                                            // count=1 (valid user D#)
  g0[1] = ldsOff;                                        // lds_addr (bytes)
  g0[2] = (u32)ga;                                       // global_addr[31:0]
  g0[3] = (u32)((ga >> 32) & 0x01FFFFFFu) | (2u << 30);  // global_addr[56:32] | type=2
  const u32 dim0  = TT * 4;                              // 256 f32 per row
  const u32 rows  = 32;
  int32x8 g1;
  g1[0] = (int)((2u << 16)      // data_size = 4B
              | (1u << 20)      // pad_enable
              | (7u << 22)      // pad_interval: every 256 dwords (1024B)
              | (3u << 25));    // pad_amount: 4 dwords (16B)
  g1[1] = (int)((dim0 & 0xFFFFu) << 16);                       // tensor_dim0 lo
  g1[2] = (int)((dim0 >> 16) | ((rows & 0xFFFFu) << 16));      // dim0 hi | tensor_dim1 lo
  g1[3] = (int)((rows >> 16) | ((dim0 & 0xFFFFu) << 16));      // dim1 hi | tile_dim0
  g1[4] = (int)(rows & 0xFFFFu);                               // tile_dim1=32, tile_dim2=0
  g1[5] = (int)(T_LEN * 4);                                    // tensor_dim0_stride lo
  g1[6] = 0;                                                   // stride0 hi | stride1 lo
  g1[7] = 0;
  int32x4 z4 = {0, 0, 0, 0};                                   // 2D: groups 2/3 unused
  int32x8 z8 = {0, 0, 0, 0, 0, 0, 0, 0};
  __builtin_amdgcn_tensor_load_to_lds(g0, g1, z4, z4, z8, 0);
}
#endif

__global__ __launch_bounds__(32) void hirnn_scan_kernel(
    const float* __restrict__ inputs, float* __restrict__ out,
    const float* INSC, const float* COEFF, const float* SQ, const float* SMSC,
    const float* SUB, const float* CRAK, const float* RecK, const float* Kr,
    const float* LG, const float* LS) {
  const int lane = threadIdx.x;        // 0..31
  const int b0 = blockIdx.x * 32;
  const int b = b0 + lane;

  Params p;
  p.insc = fminf(fmaxf(INSC[0] * 5.0f, 0.5f), 5.0f);
  p.S    = fminf(fmaxf(SMSC[0] * 500.0f, 50.0f), 500.0f);
  p.invS = 1.0f / p.S;
  p.C    = fminf(fmaxf(COEFF[0] * 400.0f, 50.0f), 400.0f);
  p.q    = fminf(fmaxf(SQ[0] * 6.0f, 0.0f), 6.0f);
  p.sub  = fminf(fmaxf(SUB[0], 0.0f), 1.0f);
  p.crak = fminf(fmaxf(CRAK[0], 0.0f), 1.0f);
  p.rk   = fminf(fmaxf(RecK[0] * 0.3f, 0.003f), 0.3f);
  p.lg   = fminf(fmaxf(LG[0] * 0.1f, 0.001f), 0.1f);
  p.kr   = fminf(fmaxf(Kr[0] * 0.1f, 0.01f), 0.1f);
  p.ls   = fminf(fmaxf(LS[0] * 10.0f, 0.01f), 10.0f);

  float SMS = 0.0f, GW = 0.0f, RS = 0.0f;
  float RS0 = 0.0f;
  float4 in0 = make_float4(0.0f, 0.0f, 0.0f, 0.0f);
  float* const orow = out + (size_t)b * T_LEN;

#if HAS_TDM
  __shared__ __align__(16) unsigned char ldsbuf[2 * BUF_BYTES];
  const u32 ldsBase = (u32)(uintptr_t)(&ldsbuf[0]);

  tdm_load_tile(inputs, b0, 0, ldsBase);                 // prefetch tile 0
  for (int tile = 0; tile < NT; ++tile) {
    if (tile + 1 < NT) {
      tdm_load_tile(inputs, b0, (tile + 1) * TT,
                    ldsBase + (u32)((tile + 1) & 1) * BUF_BYTES);
      wait_tc1();                                        // tile `tile` complete
    } else {
      wait_tc0();
    }
    asm volatile("" ::: "memory");
    const unsigned rowOff = (unsigned)(tile & 1) * BUF_BYTES + (unsigned)lane * ROW_STRIDE;
    float4 qv;
    #pragma unroll 4
    for (int dt = 0; dt < TT; ++dt) {
      float4 in = *(const float4*)(&ldsbuf[rowOff + (unsigned)dt * 16u]);
      const int t = tile * TT + dt;
      float Q = step(in, SMS, GW, RS, p);
      if (t == 0) { RS0 = RS; in0 = in; }
      ((float*)&qv)[dt & 3] = Q;
      if ((dt & 3) == 3) *(float4*)(&orow[t - 3]) = qv;  // packed 16B stores
    }
    asm volatile("" ::: "memory");
  }
#else
  const float4* __restrict__ row = (const float4*)(inputs + (u64)b * T_LEN * 4);
  float4 qv;
  for (int t = 0; t < T_LEN; ++t) {
    float4 in = row[t];
    float Q = step(in, SMS, GW, RS, p);
    if (t == 0) { RS0 = RS; in0 = in; }
    ((float*)&qv)[t & 3] = Q;
    if ((t & 3) == 3) *(float4*)(&orow[t - 3]) = qv;
  }
#endif

  // Q[0] fixup: jnp.roll wraps, so t=0 uses the FINAL SMS/GW (and post RS at t=0).
  Flux f0 = fluxes(SMS, GW, in0, p);
  orow[0] = qout(f0, RS0, in0, fast_rcp(in0.z), p);
}

extern "C" void kernel_launch(void* const* d_in, const int* in_sizes, int n_in,
                              void* d_out, int out_size, void* d_ws, size_t ws_size,
                              hipStream_t stream) {
  const float* inputs = (const float*)d_in[0];
  const float* INSC  = (const float*)d_in[1];
  const float* COEFF = (const float*)d_in[2];
  const float* SQ    = (const float*)d_in[3];
  const float* SMSC  = (const float*)d_in[4];
  const float* SUB   = (const float*)d_in[5];
  const float* CRAK  = (const float*)d_in[6];
  const float* RecK  = (const float*)d_in[7];
  const float* Kr    = (const float*)d_in[8];
  const float* LG    = (const float*)d_in[9];
  const float* LS    = (const float*)d_in[10];
  float* out = (float*)d_out;

  const int B = out_size / T_LEN;                 // reference: B=4096, T=2048
  dim3 grid((B + 31) / 32), block(32);
  hipLaunchKernelGGL(hirnn_scan_kernel, grid, block, 0, stream,
                     inputs, out, INSC, COEFF, SQ, SMSC, SUB, CRAK, RecK, Kr, LG, LS);
}